// PointPooling_18683107737686
// MI455X (gfx1250) — compile-verified
//
#include <hip/hip_runtime.h>
#include <cstddef>

typedef float v2f __attribute__((ext_vector_type(2)));
typedef float v8f __attribute__((ext_vector_type(8)));
typedef float f4v __attribute__((ext_vector_type(4)));

#define B_SEG 16
#define C_IN  64
#define C_OUT 128

// Workspace layout (floats):
//   [0, 1024)     segment sums  (16 x 64)
//   [1024, 1040)  segment counts (16)
//   [1040, 3088)  h after GEMM+BN+ReLU (16 x 128)

__global__ __launch_bounds__(256) void pp_zero_ws(float* __restrict__ ws) {
    int i = blockIdx.x * blockDim.x + threadIdx.x;
    if (i < B_SEG * C_IN + B_SEG) ws[i] = 0.0f;
}

// Pass 1: segment sums + counts. Groups of 64 threads (thread = channel),
// sorted indices -> register accumulation with flush-on-segment-change.
__global__ __launch_bounds__(256) void pp_seg_reduce(const float* __restrict__ feat,
                                                     const int* __restrict__ idx,
                                                     float* __restrict__ gsum,
                                                     float* __restrict__ gcnt,
                                                     int N, int ppg) {
    __shared__ float ssum[B_SEG * C_IN];
    __shared__ float scnt[B_SEG];
    const int tid = threadIdx.x;
    for (int i = tid; i < B_SEG * C_IN; i += 256) ssum[i] = 0.0f;
    if (tid < B_SEG) scnt[tid] = 0.0f;
    __syncthreads();

    const int  c    = tid & 63;        // channel
    const int  grp  = tid >> 6;        // 0..3 point-groups per block
    const long gid  = (long)blockIdx.x * 4 + grp;
    long start = gid * (long)ppg;
    long end   = start + (long)ppg;
    if (end > (long)N) end = (long)N;

    float acc = 0.0f, cnt = 0.0f;
    int cur = -1;
    for (long p = start; p < end; ++p) {
        int seg = idx[p];              // uniform across the group's lanes
        if (seg != cur) {
            if (cur >= 0) {
                atomicAdd(&ssum[cur * C_IN + c], acc);     // ds_add_f32
                if (c == 0) atomicAdd(&scnt[cur], cnt);
            }
            acc = 0.0f; cnt = 0.0f; cur = seg;
        }
        acc += feat[p * C_IN + c];     // 256B contiguous per group
        cnt += 1.0f;
    }
    if (cur >= 0) {
        atomicAdd(&ssum[cur * C_IN + c], acc);
        if (c == 0) atomicAdd(&scnt[cur], cnt);
    }
    __syncthreads();
    for (int i = tid; i < B_SEG * C_IN; i += 256)
        atomicAdd(&gsum[i], ssum[i]);                      // global_atomic_add_f32
    if (tid < B_SEG) atomicAdd(&gcnt[tid], scnt[tid]);
}

// Pass 2 (single block, 8 wave32s): pooled(16x64) @ W^T(64x128) via
// V_WMMA_F32_16X16X4_F32 (exact fp32), then batch-stat BN + ReLU.
__global__ __launch_bounds__(256) void pp_mm_bn_relu(const float* __restrict__ gsum,
                                                     const float* __restrict__ gcnt,
                                                     const float* __restrict__ weight,
                                                     const float* __restrict__ gamma,
                                                     const float* __restrict__ beta,
                                                     float* __restrict__ hout) {
    __shared__ float pooled[B_SEG * C_IN];   // 16 x 64
    __shared__ float hbn[B_SEG * C_OUT];     // 16 x 128
    const int tid = threadIdx.x;

    for (int i = tid; i < B_SEG * C_IN; i += 256) {
        int b = i >> 6;
        float cnt = gcnt[b];
        cnt = cnt < 1.0f ? 1.0f : cnt;
        pooled[i] = gsum[i] / cnt;
    }
    __syncthreads();

    const int wave = tid >> 5;               // 0..7 -> C_out tile [16w, 16w+16)
    const int lane = tid & 31;
    const int m  = lane & 15;                // A row (segment)
    const int kk = (lane >> 4) << 1;         // lanes 0-15: K+{0,1}; 16-31: K+{2,3}
    const int n  = (wave << 4) + (lane & 15);// B/D column (output channel)

    v8f cacc = {};                           // 16x16 f32 accumulator (8 VGPRs)
    #pragma unroll
    for (int kb = 0; kb < C_IN; kb += 4) {
        v2f a, b;
        a.x = pooled[m * C_IN + kb + kk];
        a.y = pooled[m * C_IN + kb + kk + 1];
        b.x = weight[n * C_IN + kb + kk];    // W^T[k][n] == weight[n][k]
        b.y = weight[n * C_IN + kb + kk + 1];
        cacc = __builtin_amdgcn_wmma_f32_16x16x4_f32(
            /*neg_a=*/false, a, /*neg_b=*/false, b,
            /*c_mod=*/(short)0, cacc, /*reuse_a=*/false, /*reuse_b=*/false);
    }

    #pragma unroll
    for (int r = 0; r < 8; ++r) {
        int mm = (lane < 16) ? r : (r + 8);  // C/D layout: VGPR r -> rows r / r+8
        hbn[mm * C_OUT + n] = cacc[r];
    }
    __syncthreads();

    if (tid < C_OUT) {
        float mean = 0.0f;
        #pragma unroll
        for (int mb = 0; mb < B_SEG; ++mb) mean += hbn[mb * C_OUT + tid];
        mean *= (1.0f / B_SEG);
        float var = 0.0f;
        #pragma unroll
        for (int mb = 0; mb < B_SEG; ++mb) {
            float d = hbn[mb * C_OUT + tid] - mean;
            var += d * d;
        }
        var *= (1.0f / B_SEG);                       // biased variance
        float scale = gamma[tid] * rsqrtf(var + 1e-5f);
        float shift = beta[tid] - mean * scale;
        #pragma unroll
        for (int mb = 0; mb < B_SEG; ++mb) {
            float v = hbn[mb * C_OUT + tid] * scale + shift;
            hout[mb * C_OUT + tid] = v > 0.0f ? v : 0.0f;  // ReLU
        }
    }
}

// Pass 3: one wave per point; lane l stores channels [4l,4l+4) as one float4
// -> 512B contiguous non-temporal store per wave per point.
__global__ __launch_bounds__(256) void pp_gather(const int* __restrict__ idx,
                                                 const float* __restrict__ h,
                                                 float* __restrict__ out,
                                                 int N) {
    __shared__ float hl[B_SEG * C_OUT];
    const int tid = threadIdx.x;
    for (int i = tid; i < B_SEG * C_OUT; i += 256) hl[i] = h[i];
    __syncthreads();

    const int lane   = tid & 31;
    const int wave   = (blockIdx.x * 256 + tid) >> 5;
    const int nwaves = (gridDim.x * 256) >> 5;
    for (int p = wave; p < N; p += nwaves) {
        int seg = idx[p];                                    // same for whole wave
        f4v v = *(const f4v*)&hl[seg * C_OUT + (lane << 2)]; // ds_load_b128
        f4v* dst = (f4v*)(out + (size_t)p * C_OUT) + lane;
        __builtin_nontemporal_store(v, dst);                 // NT global_store_b128
    }
}

extern "C" void kernel_launch(void* const* d_in, const int* in_sizes, int n_in,
                              void* d_out, int out_size, void* d_ws, size_t ws_size,
                              hipStream_t stream) {
    (void)n_in; (void)out_size; (void)ws_size;
    const float* feat   = (const float*)d_in[0];
    const float* weight = (const float*)d_in[1];
    const float* gamma  = (const float*)d_in[2];
    const float* beta   = (const float*)d_in[3];
    const int*   idx    = (const int*)d_in[4];
    float* out = (float*)d_out;

    float* ws   = (float*)d_ws;
    float* gsum = ws;            // 1024 floats
    float* gcnt = ws + 1024;     // 16 floats
    float* hbuf = ws + 1040;     // 2048 floats

    const int N = in_sizes[4];   // number of points (indices count)

    pp_zero_ws<<<5, 256, 0, stream>>>(ws);

    const int NB1 = 512;                       // 2048 groups of 64 threads
    const int ngroups = NB1 * 4;
    const int ppg = (N + ngroups - 1) / ngroups;
    pp_seg_reduce<<<NB1, 256, 0, stream>>>(feat, idx, gsum, gcnt, N, ppg);

    pp_mm_bn_relu<<<1, 256, 0, stream>>>(gsum, gcnt, weight, gamma, beta, hbuf);

    const int NB3 = 2048;                      // 16384 waves, ~61 points each
    pp_gather<<<NB3, 256, 0, stream>>>(idx, hbuf, out, N);
}